// TransRelation_44375602102657
// MI455X (gfx1250) — compile-verified
//
#include <hip/hip_runtime.h>

// ---------------------------------------------------------------------------
// TransRelation fused implementation for gfx1250 (MI455X), f32 WMMA path.
//   B=4, Lq=Lk=512, H=768, R=64, HD=12
// ---------------------------------------------------------------------------

typedef __attribute__((ext_vector_type(2))) float v2f;
typedef __attribute__((ext_vector_type(8))) float v8f;

constexpr int kB  = 4;
constexpr int kLQ = 512;
constexpr int kLK = 512;
constexpr int kH  = 768;
constexpr int kR  = 64;
constexpr int kHD = 12;

static __device__ __forceinline__ v8f wmma4(v2f a, v2f b, v8f c) {
  // D = A(16x4, f32) x B(4x16, f32) + C(16x16, f32); full f32 precision.
  return __builtin_amdgcn_wmma_f32_16x16x4_f32(
      /*neg_a=*/false, a, /*neg_b=*/false, b,
      /*c_mod=*/(short)0, c, /*reuse_a=*/false, /*reuse_b=*/false);
}

// ------------------------------- LayerNorm ---------------------------------
__global__ void __launch_bounds__(256)
ln_kernel(const float* __restrict__ x, const float* __restrict__ w,
          const float* __restrict__ bvec, float* __restrict__ y) {
  const int row = blockIdx.x;
  const float* xr = x + (size_t)row * kH;
  float* yr = y + (size_t)row * kH;
  __shared__ float red[32];
  const int tid = threadIdx.x;
  float s = 0.f, ss = 0.f;
  for (int c = tid; c < kH; c += 256) {
    float v = xr[c];
    s += v; ss += v * v;
  }
  for (int off = 16; off > 0; off >>= 1) {
    s  += __shfl_down(s, off, 32);
    ss += __shfl_down(ss, off, 32);
  }
  const int wave = tid >> 5;
  if ((tid & 31) == 0) { red[wave] = s; red[8 + wave] = ss; }
  __syncthreads();
  if (tid == 0) {
    float ts = 0.f, tss = 0.f;
    for (int i = 0; i < 8; ++i) { ts += red[i]; tss += red[8 + i]; }
    float mu  = ts / (float)kH;
    float var = tss / (float)kH - mu * mu;
    red[16] = mu;
    red[17] = rsqrtf(var + 1e-5f);
  }
  __syncthreads();
  const float mu = red[16], inv = red[17];
  for (int c = tid; c < kH; c += 256)
    yr[c] = (xr[c] - mu) * inv * w[c] + bvec[c];
}

// -------------------- Generic GEMM:  C = A @ W^T + bias --------------------
// A: [M,K] row-major, W: [N,K] row-major, C: [M,N]. One 16x16 C tile / wave.
__global__ void __launch_bounds__(256)
gemm_wmma_kernel(const float* __restrict__ A, const float* __restrict__ W,
                 const float* __restrict__ bias, float* __restrict__ C,
                 int M, int N, int K) {
  const int wave = threadIdx.x >> 5;
  const int lane = threadIdx.x & 31;
  const int ntiles = N >> 4;
  const int tile = blockIdx.x * 8 + wave;
  if (tile >= (M >> 4) * ntiles) return;
  const int mt = tile / ntiles;
  const int nt = tile % ntiles;
  const int half = lane >> 4;   // lanes 16-31 hold K+2..K+3 per ISA layout
  const int lr   = lane & 15;

  const float* arow = A + (size_t)(mt * 16 + lr) * K + half * 2;
  const float* brow = W + (size_t)(nt * 16 + lr) * K + half * 2;

  v8f acc = {};
  for (int k = 0; k < K; k += 4) {
    v2f a = *(const v2f*)(arow + k);
    v2f b = *(const v2f*)(brow + k);
    acc = wmma4(a, b, acc);
  }
  const int col = nt * 16 + lr;
  const float bb = bias ? bias[col] : 0.f;
#pragma unroll
  for (int v = 0; v < 8; ++v) {
    const int r = mt * 16 + v + half * 8;   // C layout: VGPR v -> M=v (+8 hi half)
    C[(size_t)r * N + col] = acc[v] + bb;
  }
}

// --------------------------- Fused relation attn ---------------------------
// Block = (batch b, 16-query tile it). 256 threads = 8 waves.
// For each 16-wide j chunk:
//   Stage A (all 8 waves): mixed logits L[jj,r'] = sum_r (q_i[r]*k_jj[r])*Wt[r',r]
//     via WMMA f32 16x16x4 (K-dim r = 64 -> 16 steps); exp(+mask) -> LDS P.
//   Stage B (4 waves per r'-half): O[i,(d|l)] += P[i,jj] @ Vaug[jj,(d|1)]
//     accumulators live in registers across all 32 chunks; col 12 of Vaug
//     is a ones column so the softmax denominator comes out of the WMMA.
__global__ void __launch_bounds__(256)
attn_kernel(const float* __restrict__ qm, const float* __restrict__ km,
            const float* __restrict__ vm, const float* __restrict__ Wt,
            const float* __restrict__ mask, float* __restrict__ omid) {
  const int b    = blockIdx.x >> 5;
  const int it   = blockIdx.x & 31;
  const int tid  = threadIdx.x;
  const int wave = tid >> 5;
  const int lane = tid & 31;
  const int half = lane >> 4;
  const int lr   = lane & 15;

  __shared__ __align__(16) float sQ[16 * 64];     // [i][r]
  __shared__ __align__(16) float sWt[64 * 64];    // [r'][r] (as given)
  __shared__ __align__(16) float sK[16 * 64];     // [jj][r]
  __shared__ __align__(16) float sMask[16 * 16];  // [i][jj]
  __shared__ __align__(16) float sP[16 * 529];    // [i][jj*33 + rpl], pad-free banks
  __shared__ __align__(16) float sRed[8 * 16];    // per-wave denom broadcast

  for (int idx = tid; idx < 16 * 64; idx += 256)
    sQ[idx] = qm[(size_t)(b * kLQ + it * 16 + (idx >> 6)) * kR + (idx & 63)];
  for (int idx = tid; idx < 64 * 64; idx += 256)
    sWt[idx] = Wt[idx];

  v8f acc[8];
#pragma unroll
  for (int t = 0; t < 8; ++t) acc[t] = v8f{};

  __syncthreads();

  for (int ch = 0; ch < kLK / 16; ++ch) {
    const int j0 = ch * 16;
    __syncthreads();
    for (int idx = tid; idx < 16 * 64; idx += 256)
      sK[idx] = km[(size_t)(b * kLK + j0 + (idx >> 6)) * kR + (idx & 63)];
    sMask[tid] = mask[(size_t)(b * kLQ + it * 16 + (tid >> 4)) * kLK + j0 + (tid & 15)];
    __syncthreads();

    for (int hh = 0; hh < 2; ++hh) {   // r' halves: [0,32) and [32,64)
      // ---- Stage A: mixed logits -> exp -> sP -----------------------------
#pragma unroll
      for (int t = 0; t < 4; ++t) {
        const int job = wave * 4 + t;          // 32 jobs: (i, local ntile)
        const int i   = job >> 1;
        const int ntl = job & 1;
        const int rpb = hh * 32 + ntl * 16;    // r' base of this 16-wide tile
        const float* qrow = sQ + i * 64 + half * 2;
        const float* krow = sK + lr * 64 + half * 2;           // A row M = jj = lr
        const float* wrow = sWt + (rpb + lr) * 64 + half * 2;  // B col N = r' = lr
        v8f c = {};
#pragma unroll
        for (int k = 0; k < 64; k += 4) {
          v2f qa = *(const v2f*)(qrow + k);
          v2f ka = *(const v2f*)(krow + k);
          v2f a  = qa * ka;                    // S[jj,r] = q_i[r]*k_jj[r]
          v2f bb = *(const v2f*)(wrow + k);
          c = wmma4(a, bb, c);
        }
        const int rpl = ntl * 16 + lr;         // local r' in [0,32)
#pragma unroll
        for (int v = 0; v < 8; ++v) {
          const int jj = v + half * 8;
          sP[i * 529 + jj * 33 + rpl] = __expf(c[v] + sMask[i * 16 + jj]);
        }
      }
      __syncthreads();
      // ---- Stage B: accumulate P @ Vaug -----------------------------------
      if ((wave >> 2) == hh) {
#pragma unroll
        for (int t = 0; t < 8; ++t) {
          const int rp  = wave * 8 + t;        // global r' owned by this wave
          const int rpl = rp - hh * 32;
          const size_t vbase = (size_t)(b * kLK + j0) * kH + rp * kHD;
#pragma unroll
          for (int k = 0; k < 16; k += 4) {    // K-dim = jj
            const int kj = k + half * 2;
            v2f a;                              // A row M = i = lr
            a.x = sP[lr * 529 + kj * 33 + rpl];
            a.y = sP[lr * 529 + (kj + 1) * 33 + rpl];
            v2f bb;                             // B col N = d = lr (aug col 12 = 1)
            if (lr < kHD) {
              bb.x = vm[vbase + (size_t)kj * kH + lr];
              bb.y = vm[vbase + (size_t)(kj + 1) * kH + lr];
            } else if (lr == kHD) {
              bb.x = 1.f; bb.y = 1.f;
            } else {
              bb.x = 0.f; bb.y = 0.f;
            }
            acc[t] = wmma4(a, bb, acc[t]);
          }
        }
      }
      __syncthreads();
    }
  }

  // ---- Epilogue: normalize by denominator (accumulator column 12) ---------
#pragma unroll
  for (int t = 0; t < 8; ++t) {
    const int rp = wave * 8 + t;
    if (lr == 12) {                     // lanes 12 & 28 hold the ones-column
#pragma unroll
      for (int v = 0; v < 8; ++v) sRed[wave * 16 + v + half * 8] = acc[t][v];
    }
    __syncthreads();
    if (lr < kHD) {
#pragma unroll
      for (int v = 0; v < 8; ++v) {
        const int i = v + half * 8;
        const float l = sRed[wave * 16 + i];
        omid[(size_t)(b * kLQ + it * 16 + i) * kH + rp * kHD + lr] = acc[t][v] / l;
      }
    }
    __syncthreads();
  }
}

// ------------------------------- launcher ----------------------------------
extern "C" void kernel_launch(void* const* d_in, const int* in_sizes, int n_in,
                              void* d_out, int out_size, void* d_ws, size_t ws_size,
                              hipStream_t stream) {
  (void)in_sizes; (void)n_in; (void)out_size; (void)ws_size;
  const float* tgt  = (const float*)d_in[0];
  const float* mem  = (const float*)d_in[1];
  const float* mask = (const float*)d_in[2];
  const float* lnqw = (const float*)d_in[3];
  const float* lnqb = (const float*)d_in[4];
  const float* lnkw = (const float*)d_in[5];
  const float* lnkb = (const float*)d_in[6];
  const float* Wq   = (const float*)d_in[7];
  const float* bq   = (const float*)d_in[8];
  const float* Wk   = (const float*)d_in[9];
  const float* bk   = (const float*)d_in[10];
  const float* Wv   = (const float*)d_in[11];
  const float* bv   = (const float*)d_in[12];
  const float* Wt   = (const float*)d_in[13];
  const float* Wo   = (const float*)d_in[14];
  const float* bo   = (const float*)d_in[15];
  float* out = (float*)d_out;

  float* ws = (float*)d_ws;
  const size_t RT = (size_t)kB * kLQ;      // 2048 rows
  float* t_n  = ws;                        // [2048,768] normalized tgt
  float* m_n  = t_n + RT * kH;             // [2048,768] normalized mem
  float* qp   = m_n + RT * kH;             // [2048,64]
  float* kp   = qp + RT * kR;              // [2048,64]
  float* vp   = kp + RT * kR;              // [2048,768]
  float* omid = vp + RT * kH;              // [2048,768] attention output

  ln_kernel<<<(unsigned)RT, 256, 0, stream>>>(tgt, lnqw, lnqb, t_n);
  ln_kernel<<<(unsigned)RT, 256, 0, stream>>>(mem, lnkw, lnkb, m_n);

  auto launch_gemm = [&](const float* A, const float* W, const float* bias,
                         float* C, int M, int N, int K) {
    const int tiles = (M / 16) * (N / 16);
    gemm_wmma_kernel<<<(tiles + 7) / 8, 256, 0, stream>>>(A, W, bias, C, M, N, K);
  };
  launch_gemm(t_n, Wq, bq, qp, kB * kLQ, kR, kH);   // q
  launch_gemm(m_n, Wk, bk, kp, kB * kLK, kR, kH);   // k
  launch_gemm(m_n, Wv, bv, vp, kB * kLK, kH, kH);   // v

  attn_kernel<<<kB * (kLQ / 16), 256, 0, stream>>>(qp, kp, vp, Wt, mask, omid);

  launch_gemm(omid, Wo, bo, out, kB * kLQ, kH, kH); // final projection
}